// GCNLayer_60885456388921
// MI455X (gfx1250) — compile-verified
//
#include <hip/hip_runtime.h>
#include <math.h>

#define N_NODES 50000
#define N_EDGES 800000
#define C_IN    128
#define C_OUT   128
#define HEADS   2
#define NEG_SLOPE 0.2f

typedef __attribute__((ext_vector_type(2))) float v2f;
typedef __attribute__((ext_vector_type(8))) float v8f;

__device__ __forceinline__ float wave_reduce_sum(float v) {
#pragma unroll
  for (int m = 16; m >= 1; m >>= 1) v += __shfl_xor(v, m, 32);
  return v;
}

__device__ __forceinline__ float dot4(float4 a, float4 b) {
  return a.x * b.x + a.y * b.y + a.z * b.z + a.w * b.w;
}

// Order-preserving encoding of float into uint for atomicMax.
__device__ __forceinline__ unsigned float_order(float f) {
  unsigned u = __float_as_uint(f);
  return (u & 0x80000000u) ? ~u : (u | 0x80000000u);
}
__device__ __forceinline__ float order_float(unsigned u) {
  return (u & 0x80000000u) ? __uint_as_float(u ^ 0x80000000u) : __uint_as_float(~u);
}

// ---------------- 1. LayerNorm + ReLU : one wave per node ----------------
__global__ void __launch_bounds__(256)
ln_relu_kernel(const float* __restrict__ x, const float* __restrict__ gamma,
               const float* __restrict__ beta, float* __restrict__ h) {
  int node = (int)((blockIdx.x * blockDim.x + threadIdx.x) >> 5);
  int lane = threadIdx.x & 31;
  if (node >= N_NODES) return;
  float4 xv = ((const float4*)(x + (size_t)node * C_IN))[lane];
  float s  = xv.x + xv.y + xv.z + xv.w;
  float sq = xv.x * xv.x + xv.y * xv.y + xv.z * xv.z + xv.w * xv.w;
  s  = wave_reduce_sum(s);
  sq = wave_reduce_sum(sq);
  float mu  = s * (1.0f / C_IN);
  float var = sq * (1.0f / C_IN) - mu * mu;
  float inv = rsqrtf(var + 1e-5f);
  float4 gv = ((const float4*)gamma)[lane];
  float4 bv = ((const float4*)beta)[lane];
  float4 hv;
  hv.x = fmaxf(0.f, (xv.x - mu) * inv * gv.x + bv.x);
  hv.y = fmaxf(0.f, (xv.y - mu) * inv * gv.y + bv.y);
  hv.z = fmaxf(0.f, (xv.z - mu) * inv * gv.z + bv.z);
  hv.w = fmaxf(0.f, (xv.w - mu) * inv * gv.w + bv.w);
  ((float4*)(h + (size_t)node * C_IN))[lane] = hv;
}

// ---------------- 2. GEMM h[50000x128] @ W[128x256] via V_WMMA_F32_16X16X4_F32
// Block = 256 threads = 8 waves per 16-row strip; each wave -> two 16x16 tiles.
__global__ void __launch_bounds__(256)
gemm_wmma_kernel(const float* __restrict__ h, const float* __restrict__ W,
                 float* __restrict__ xw) {
  const int lane   = threadIdx.x & 31;
  const int waveId = threadIdx.x >> 5;
  const int m  = lane & 15;        // row within A-tile / col within B-tile
  const int hi = lane >> 4;        // K-pair selector (0: K=0,1 ; 1: K=2,3)
  const int rowBase = blockIdx.x * 16;
  const int col0 = (waveId * 2) * 16;
  const int col1 = col0 + 16;

  v8f c0 = {};
  v8f c1 = {};
  const float* hrow = h + (size_t)(rowBase + m) * C_IN + 2 * hi;

#pragma unroll
  for (int k = 0; k < C_IN; k += 4) {
    float2 av = *(const float2*)(hrow + k);   // A: 16x4 f32 fragment (2 VGPRs)
    v2f a; a.x = av.x; a.y = av.y;
    const float* wp = W + (size_t)(k + 2 * hi) * (HEADS * C_OUT);
    v2f b0, b1;                               // B: 4x16 f32 fragments
    b0.x = wp[col0 + m];  b0.y = wp[HEADS * C_OUT + col0 + m];
    b1.x = wp[col1 + m];  b1.y = wp[HEADS * C_OUT + col1 + m];
    c0 = __builtin_amdgcn_wmma_f32_16x16x4_f32(false, a, false, b0, (short)0, c0, false, false);
    c1 = __builtin_amdgcn_wmma_f32_16x16x4_f32(false, a, false, b1, (short)0, c1, false, false);
  }

#pragma unroll
  for (int r = 0; r < 8; ++r) {               // D layout: VGPR r -> M = r + 8*hi
    size_t row = (size_t)(rowBase + r + 8 * hi) * (HEADS * C_OUT);
    xw[row + col0 + m] = c0[r];
    xw[row + col1 + m] = c1[r];
  }
}

// ---------------- 3. Attention scores: one wave per node (both heads) ------
__global__ void __launch_bounds__(256)
att_kernel(const float* __restrict__ xw, const float* __restrict__ att_src,
           const float* __restrict__ att_dst, float* __restrict__ a_src,
           float* __restrict__ a_dst) {
  int node = (int)((blockIdx.x * blockDim.x + threadIdx.x) >> 5);
  int lane = threadIdx.x & 31;
  if (node >= N_NODES) return;
  const float4* row = (const float4*)(xw + (size_t)node * (HEADS * C_OUT));
  float4 x0 = row[lane];
  float4 x1 = row[32 + lane];
  float4 s0 = ((const float4*)att_src)[lane];
  float4 s1 = ((const float4*)att_src)[32 + lane];
  float4 d0 = ((const float4*)att_dst)[lane];
  float4 d1 = ((const float4*)att_dst)[32 + lane];
  float as0 = wave_reduce_sum(dot4(x0, s0));
  float as1 = wave_reduce_sum(dot4(x1, s1));
  float ad0 = wave_reduce_sum(dot4(x0, d0));
  float ad1 = wave_reduce_sum(dot4(x1, d1));
  if (lane == 0) {
    a_src[node * 2 + 0] = as0;
    a_src[node * 2 + 1] = as1;
    a_dst[node * 2 + 0] = ad0;
    a_dst[node * 2 + 1] = ad1;
  }
}

// ---------------- 4. Init accumulators -------------------------------------
__global__ void __launch_bounds__(256)
init_kernel(unsigned* __restrict__ m_enc, float* __restrict__ s,
            float* __restrict__ acc) {
  int i = blockIdx.x * blockDim.x + threadIdx.x;
  if (i < N_NODES * HEADS) { m_enc[i] = 0x007FFFFFu; /* encode(-inf) */ s[i] = 0.f; }
  if (i < N_NODES * C_OUT) acc[i] = 0.f;
}

// ---------------- 5. Edge: alpha + segment max -----------------------------
__global__ void __launch_bounds__(256)
edge_alpha_max_kernel(const int* __restrict__ ei, const float* __restrict__ a_src,
                      const float* __restrict__ a_dst, float* __restrict__ alpha,
                      unsigned* __restrict__ m_enc) {
  int e = blockIdx.x * blockDim.x + threadIdx.x;
  if (e >= N_EDGES) return;
  int src = ei[e], dst = ei[N_EDGES + e];
#pragma unroll
  for (int hh = 0; hh < HEADS; ++hh) {
    float al = a_src[src * 2 + hh] + a_dst[dst * 2 + hh];
    al = (al > 0.f) ? al : NEG_SLOPE * al;
    alpha[(size_t)e * 2 + hh] = al;
    atomicMax(&m_enc[dst * 2 + hh], float_order(al));
  }
}

// ---------------- 6. Edge: exp + segment sum -------------------------------
__global__ void __launch_bounds__(256)
edge_exp_sum_kernel(const int* __restrict__ ei, const unsigned* __restrict__ m_enc,
                    float* __restrict__ alpha, float* __restrict__ s) {
  int e = blockIdx.x * blockDim.x + threadIdx.x;
  if (e >= N_EDGES) return;
  int dst = ei[N_EDGES + e];
#pragma unroll
  for (int hh = 0; hh < HEADS; ++hh) {
    float mv = order_float(m_enc[dst * 2 + hh]);
    if (!isfinite(mv)) mv = 0.f;
    float ev = expf(alpha[(size_t)e * 2 + hh] - mv);
    alpha[(size_t)e * 2 + hh] = ev;   // reuse buffer: now holds exp value
    atomicAdd(&s[dst * 2 + hh], ev);
  }
}

// ---------------- 7. Edge scatter: one wave per (edge, head) ---------------
// Head mean folded into coef (0.5x); both heads accumulate into acc[N,128].
__global__ void __launch_bounds__(256)
edge_scatter_kernel(const int* __restrict__ ei, const float* __restrict__ xw,
                    const float* __restrict__ alpha, const float* __restrict__ s,
                    float* __restrict__ acc) {
  long long idx = ((long long)blockIdx.x * blockDim.x + threadIdx.x) >> 5;
  int lane = threadIdx.x & 31;
  if (idx >= (long long)N_EDGES * HEADS) return;
  int e  = (int)(idx >> 1);
  int hh = (int)(idx & 1);
  int src = ei[e], dst = ei[N_EDGES + e];
  float ev   = alpha[(size_t)e * 2 + hh];
  float sv   = s[dst * 2 + hh];
  float coef = 0.5f * ev / (sv + 1e-16f);
  float4 xv = ((const float4*)(xw + (size_t)src * (HEADS * C_OUT) + hh * C_OUT))[lane];
  float* ap = acc + (size_t)dst * C_OUT + lane * 4;
  atomicAdd(ap + 0, xv.x * coef);
  atomicAdd(ap + 1, xv.y * coef);
  atomicAdd(ap + 2, xv.z * coef);
  atomicAdd(ap + 3, xv.w * coef);
}

// ---------------- 8. Finalize: residual + bias -----------------------------
__global__ void __launch_bounds__(256)
finalize_kernel(const float* __restrict__ x, const float* __restrict__ acc,
                const float* __restrict__ bias, float* __restrict__ out) {
  int i = blockIdx.x * blockDim.x + threadIdx.x;
  if (i >= N_NODES * C_OUT) return;
  out[i] = x[i] + acc[i] + bias[i & (C_OUT - 1)];
}

extern "C" void kernel_launch(void* const* d_in, const int* in_sizes, int n_in,
                              void* d_out, int out_size, void* d_ws, size_t ws_size,
                              hipStream_t stream) {
  const float* x        = (const float*)d_in[0];
  const int*   ei       = (const int*)d_in[1];   // [2, E]
  const float* W        = (const float*)d_in[2];
  const float* att_src  = (const float*)d_in[3];
  const float* att_dst  = (const float*)d_in[4];
  const float* bias     = (const float*)d_in[5];
  const float* ln_gamma = (const float*)d_in[6];
  const float* ln_beta  = (const float*)d_in[7];
  float* out = (float*)d_out;

  char* ws = (char*)d_ws;
  float*    h     = (float*)ws;    ws += (size_t)N_NODES * C_IN * sizeof(float);
  float*    xw    = (float*)ws;    ws += (size_t)N_NODES * HEADS * C_OUT * sizeof(float);
  float*    a_src = (float*)ws;    ws += (size_t)N_NODES * HEADS * sizeof(float);
  float*    a_dst = (float*)ws;    ws += (size_t)N_NODES * HEADS * sizeof(float);
  unsigned* m_enc = (unsigned*)ws; ws += (size_t)N_NODES * HEADS * sizeof(unsigned);
  float*    ssum  = (float*)ws;    ws += (size_t)N_NODES * HEADS * sizeof(float);
  float*    alpha = (float*)ws;    ws += (size_t)N_EDGES * HEADS * sizeof(float);
  float*    acc   = (float*)ws;    ws += (size_t)N_NODES * C_OUT * sizeof(float);

  const int B = 256;
  // 1. LayerNorm + ReLU (8 waves/block -> 8 nodes/block)
  ln_relu_kernel<<<(N_NODES + 7) / 8, B, 0, stream>>>(x, ln_gamma, ln_beta, h);
  // 2. WMMA GEMM: 3125 blocks of 16 rows
  gemm_wmma_kernel<<<N_NODES / 16, B, 0, stream>>>(h, W, xw);
  // 3. Attention scores
  att_kernel<<<(N_NODES + 7) / 8, B, 0, stream>>>(xw, att_src, att_dst, a_src, a_dst);
  // 4. Init segment buffers
  init_kernel<<<(N_NODES * C_OUT + B - 1) / B, B, 0, stream>>>(m_enc, ssum, acc);
  // 5. alpha + segment max
  edge_alpha_max_kernel<<<(N_EDGES + B - 1) / B, B, 0, stream>>>(ei, a_src, a_dst, alpha, m_enc);
  // 6. exp + segment sum
  edge_exp_sum_kernel<<<(N_EDGES + B - 1) / B, B, 0, stream>>>(ei, m_enc, alpha, ssum);
  // 7. gather/scatter messages (1 wave per edge-head)
  long long nwaves = (long long)N_EDGES * HEADS;
  edge_scatter_kernel<<<(unsigned)((nwaves + 7) / 8), B, 0, stream>>>(ei, xw, alpha, ssum, acc);
  // 8. residual + bias
  finalize_kernel<<<(N_NODES * C_OUT + B - 1) / B, B, 0, stream>>>(x, acc, bias, out);
}